// BondLengthConstraint_61048665145611
// MI455X (gfx1250) — compile-verified
//
#include <hip/hip_runtime.h>

#define CONSTRAINT_WEIGHT 0.1f
#define CORRECTION_FACTOR 0.1f
#define BLOCK 256
#define WAVES_PER_BLOCK (BLOCK / 32)   // wave32 on CDNA5

// ---------------------------------------------------------------------------
// Kernel 1: pos_new = pos  (d_out poisoned by harness; rewrite each call)
// ---------------------------------------------------------------------------
__global__ void blc_init_kernel(const float* __restrict__ pos,
                                float* __restrict__ out,
                                int n) {
    int i = blockIdx.x * blockDim.x + threadIdx.x;
    if (i < n) out[i] = pos[i];
}

// ---------------------------------------------------------------------------
// Kernel 2: one thread per edge.
//   NT-load edge streams (192MB streamed once -> don't evict L2-resident pos),
//   gather endpoints, compute half-adjustment, scatter with HW f32 atomics.
//   Loss: wave32 shfl reduce -> LDS block reduce -> plain store of one
//   partial per block (no global atomics, deterministic).
// ---------------------------------------------------------------------------
__global__ void blc_edge_kernel(const float* __restrict__ pos,
                                const long long* __restrict__ edge_index,
                                const long long* __restrict__ bond_types,
                                const float* __restrict__ target_lengths,
                                const float* __restrict__ length_adjustment,
                                float* __restrict__ out,
                                float* __restrict__ block_partials,
                                int n_edges) {
    __shared__ float s_tgt[5];
    __shared__ float s_part[WAVES_PER_BLOCK];
    if (threadIdx.x < 5)
        s_tgt[threadIdx.x] = target_lengths[threadIdx.x] + length_adjustment[threadIdx.x];
    __syncthreads();

    int e = blockIdx.x * blockDim.x + threadIdx.x;
    float err = 0.0f;

    if (e < n_edges) {
        // Streamed-once 192MB: non-temporal so pos/out stay hot in 192MB L2.
        long long row = __builtin_nontemporal_load(&edge_index[e]);
        long long col = __builtin_nontemporal_load(&edge_index[(long long)n_edges + e]);
        long long bt  = __builtin_nontemporal_load(&bond_types[e]);
        int bti = (int)(bt < 0 ? 0 : (bt > 4 ? 4 : bt));

        const float* pr = pos + 3ll * row;   // 12B gathers; L2-resident (6MB)
        const float* pc = pos + 3ll * col;
        float dx = pr[0] - pc[0];
        float dy = pr[1] - pc[1];
        float dz = pr[2] - pc[2];

        float len = sqrtf(dx * dx + dy * dy + dz * dz);
        len = fmaxf(len, 1e-6f);

        float tgt = s_tgt[bti];
        err = fabsf(len - tgt);

        float ratio = fminf(fmaxf(tgt / len, 0.8f), 1.2f);
        float s = (ratio - 1.0f) * (CORRECTION_FACTOR * 0.5f);
        float hx = dx * s, hy = dy * s, hz = dz * s;

        float* orow = out + 3ll * row;
        float* ocol = out + 3ll * col;
        // native global_atomic_add_f32; duplicates accumulate like .at[].add
        unsafeAtomicAdd(&orow[0],  hx);
        unsafeAtomicAdd(&orow[1],  hy);
        unsafeAtomicAdd(&orow[2],  hz);
        unsafeAtomicAdd(&ocol[0], -hx);
        unsafeAtomicAdd(&ocol[1], -hy);
        unsafeAtomicAdd(&ocol[2], -hz);
    }

    // wave32 butterfly reduction (all lanes execute; tail lanes contribute 0)
    for (int off = 16; off > 0; off >>= 1)
        err += __shfl_xor(err, off, 32);

    if ((threadIdx.x & 31) == 0)
        s_part[threadIdx.x >> 5] = err;
    __syncthreads();

    if (threadIdx.x == 0) {
        float bsum = 0.0f;
        #pragma unroll
        for (int w = 0; w < WAVES_PER_BLOCK; ++w) bsum += s_part[w];
        block_partials[blockIdx.x] = bsum;     // plain store, zero contention
    }
}

// ---------------------------------------------------------------------------
// Kernel 3: single block sums all block partials in f64, writes final loss.
// ---------------------------------------------------------------------------
__global__ void blc_reduce_kernel(const float* __restrict__ partials,
                                  int n_parts,
                                  float* __restrict__ out_loss,
                                  double inv_e) {
    __shared__ double s_acc[32];
    double acc = 0.0;
    for (int i = threadIdx.x; i < n_parts; i += blockDim.x)
        acc += (double)partials[i];

    for (int off = 16; off > 0; off >>= 1)
        acc += __shfl_xor(acc, off, 32);

    int wave = threadIdx.x >> 5;
    int lane = threadIdx.x & 31;
    if (lane == 0) s_acc[wave] = acc;
    __syncthreads();

    if (threadIdx.x == 0) {
        double total = 0.0;
        int n_waves = blockDim.x >> 5;
        for (int w = 0; w < n_waves; ++w) total += s_acc[w];
        *out_loss = (float)(total * inv_e * (double)CONSTRAINT_WEIGHT);
    }
}

extern "C" void kernel_launch(void* const* d_in, const int* in_sizes, int n_in,
                              void* d_out, int out_size, void* d_ws, size_t ws_size,
                              hipStream_t stream) {
    const float*     pos      = (const float*)d_in[0];      // [N,3] f32
    const long long* edge_idx = (const long long*)d_in[1];  // [2,E] i64
    const long long* bond_ty  = (const long long*)d_in[2];  // [E]   i64
    const float*     tgt_len  = (const float*)d_in[3];      // [5]   f32
    const float*     len_adj  = (const float*)d_in[4];      // [5]   f32

    float* out = (float*)d_out;            // [N*3] pos_new, then [1] loss
    float* partials = (float*)d_ws;        // one float per edge-block

    const int n_pos   = in_sizes[0];       // 1,500,000
    const int n_edges = in_sizes[2];       // 8,000,000

    const int n_edge_blocks = (n_edges + BLOCK - 1) / BLOCK;   // 31250

    blc_init_kernel<<<(n_pos + BLOCK - 1) / BLOCK, BLOCK, 0, stream>>>(
        pos, out, n_pos);

    blc_edge_kernel<<<n_edge_blocks, BLOCK, 0, stream>>>(
        pos, edge_idx, bond_ty, tgt_len, len_adj, out, partials, n_edges);

    blc_reduce_kernel<<<1, 1024, 0, stream>>>(
        partials, n_edge_blocks, out + n_pos, 1.0 / (double)n_edges);
}